// Decoder_70884140253630
// MI455X (gfx1250) — compile-verified
//
#include <hip/hip_runtime.h>
#include <hip/hip_bf16.h>
#include <math.h>

typedef float v2f __attribute__((ext_vector_type(2)));
typedef float v8f __attribute__((ext_vector_type(8)));

#define BB 512
#define SS 128
#define DD 128
#define NH 8
#define DKH 16
#define D3 384
#define NP 512      // fused projection width: [gK | gV | encUpd | M]
#define NSTEPS 126

#define NEGINF (-__builtin_inff())

// ---------------------------------------------------------------------------
// Kernel 0a: build fused weight Wbig[128][512] =
//   [ W_nodes[:,0:128] | W_nodes[:,128:256] | W_upd[0:128,:] | Wcomb ]
// where Wcomb[k][j] = sum_d W_nodes[k][256+d] * W_out[j][d]
// ---------------------------------------------------------------------------
__global__ void wbig_kernel(const float* __restrict__ Wn,
                            const float* __restrict__ Wu,
                            const float* __restrict__ Wo,
                            float* __restrict__ Wbig) {
    int k = blockIdx.x;   // 0..127
    int j = threadIdx.x;  // 0..127
    Wbig[(size_t)k * NP + j]            = Wn[(size_t)k * D3 + j];          // gK cols
    Wbig[(size_t)k * NP + DD + j]       = Wn[(size_t)k * D3 + DD + j];     // gV cols
    Wbig[(size_t)k * NP + 2 * DD + j]   = Wu[(size_t)k * DD + j];          // encUpd cols
    const float* ln = Wn + (size_t)k * D3 + 2 * DD;                        // logK weights
    float acc = 0.f;
    for (int d = 0; d < DD; ++d) acc += ln[d] * Wo[(size_t)j * DD + d];
    Wbig[(size_t)k * NP + 3 * DD + j] = acc;                               // Wcomb
}

// ---------------------------------------------------------------------------
// Kernel 0b: step-invariant projection vectors
//   vtProj = vtmax @ W_upd[128:256], vmProj = vm @ W_upd[256:384],
//   phUpd  = W_placeholder @ W_upd[0:128]
// ---------------------------------------------------------------------------
__global__ void vecs_kernel(const float* __restrict__ Wu,
                            const float* __restrict__ vt,
                            const float* __restrict__ vmv,
                            const float* __restrict__ ph,
                            float* __restrict__ out3) {
    int t = threadIdx.x; // 128
    float a = 0.f, b2 = 0.f, c = 0.f;
    for (int k = 0; k < DD; ++k) {
        a  += vt[k]  * Wu[(size_t)(DD + k) * DD + t];
        b2 += vmv[k] * Wu[(size_t)(2 * DD + k) * DD + t];
        c  += ph[k]  * Wu[(size_t)k * DD + t];
    }
    out3[t] = a; out3[DD + t] = b2; out3[2 * DD + t] = c;
}

// ---------------------------------------------------------------------------
// Kernel 1: pairwise distances + depot distances. One block per batch.
// ---------------------------------------------------------------------------
__global__ void geom_kernel(const float* __restrict__ loc,
                            float* __restrict__ dists,
                            float* __restrict__ d2depot) {
    int b = blockIdx.x;
    int t = threadIdx.x; // 128
    __shared__ float lx[SS], ly[SS];
    lx[t] = loc[(b * SS + t) * 2 + 0];
    ly[t] = loc[(b * SS + t) * 2 + 1];
    __syncthreads();
    float xi = lx[t], yi = ly[t];
    float dx0 = xi - lx[0], dy0 = yi - ly[0];
    d2depot[b * SS + t] = sqrtf(dx0 * dx0 + dy0 * dy0 + 1e-12f);
    for (int j = 0; j < SS; ++j) {
        float dx = xi - lx[j], dy = yi - ly[j];
        dists[((size_t)b * SS + t) * SS + j] = sqrtf(dx * dx + dy * dy + 1e-12f);
    }
}

// ---------------------------------------------------------------------------
// Kernel 2: h_hat = mean(enc, axis=S) @ W_ctx. One block per batch.
// ---------------------------------------------------------------------------
__global__ void hhat_kernel(const float* __restrict__ enc,
                            const float* __restrict__ W_ctx,
                            float* __restrict__ h_hat) {
    int b = blockIdx.x;
    int t = threadIdx.x; // 128
    __shared__ float mean[DD];
    float acc = 0.f;
    for (int s = 0; s < SS; ++s) acc += enc[((size_t)b * SS + s) * DD + t];
    mean[t] = acc * (1.0f / SS);
    __syncthreads();
    float o = 0.f;
    for (int k = 0; k < DD; ++k) o += mean[k] * W_ctx[k * DD + t];
    h_hat[b * DD + t] = o;
}

// ---------------------------------------------------------------------------
// Kernel 3: P = enc(view [M=B*S,128]) @ Wbig[128,512] via fp32 WMMA.
// One wave32 per 16x16 C tile; K-loop of 32 x V_WMMA_F32_16X16X4_F32.
// Block = 256 threads (8 waves), grid = (M/16, 512/16/8) = (4096, 4).
// ---------------------------------------------------------------------------
__global__ void proj_wmma_kernel(const float* __restrict__ A,
                                 const float* __restrict__ Wb,
                                 float* __restrict__ C) {
    int lane = threadIdx.x & 31;
    int wave = threadIdx.x >> 5;               // 0..7
    int mtile = blockIdx.x;                    // 0..4095
    int ntile = blockIdx.y * 8 + wave;         // 0..31
    int half = lane >> 4;                      // K plane pair select
    int l15  = lane & 15;
    int mrow = mtile * 16 + l15;
    int ncol = ntile * 16 + l15;

    v8f c = {};
    const float* arow = A + (size_t)mrow * DD;
    for (int kk = 0; kk < DD; kk += 4) {
        int ka = kk + half * 2;                // lanes16-31 hold K+2,K+3 planes
        v2f a, bm;
        a.x  = arow[ka];
        a.y  = arow[ka + 1];
        bm.x = Wb[(size_t)ka * NP + ncol];
        bm.y = Wb[(size_t)(ka + 1) * NP + ncol];
        c = __builtin_amdgcn_wmma_f32_16x16x4_f32(
                false, a, false, bm, (short)0, c, false, false);
    }
#pragma unroll
    for (int r = 0; r < 8; ++r) {
        int row = mtile * 16 + r + half * 8;
        C[(size_t)row * NP + ncol] = c[r];
    }
}

// ---------------------------------------------------------------------------
// Kernel 4: fused 126-step greedy decode. One block (128 thr) per batch elem.
// P row layout per (b,s): [gK(128) | gV(128) | encUpd(128) | M(128)]
// query = h_hat + encUpd[last] + Tm*vtProj + m*vmProj   (GEMV hoisted)
// logits[s] = heads . M[s]                              (glimpse fused)
// ---------------------------------------------------------------------------
__global__ void decode_kernel(const float* __restrict__ scores,
                              const float* __restrict__ Tmax_in,
                              const int*   __restrict__ m_in,
                              const float* __restrict__ P,
                              const float* __restrict__ vecs,   // vtProj|vmProj|phUpd
                              const float* __restrict__ dists,
                              const float* __restrict__ d2depot,
                              const float* __restrict__ h_hat_g,
                              float* __restrict__ out_logp,
                              float* __restrict__ out_total,
                              float* __restrict__ out_sol) {
    int b = blockIdx.x;
    int t = threadIdx.x; // 128

    __shared__ float q[DD];
    __shared__ float att[NH][SS];
    __shared__ float headsArr[DD];
    __shared__ float logits[SS];
    __shared__ float red[128];
    __shared__ int   redi[128];
    __shared__ int   maskSh[SS];
    __shared__ float hhat[DD], vtP[DD], vmP[DD], phU[DD];
    __shared__ float sTm, sTmax0, sTotal, sLogp;
    __shared__ int   sM, sLast, sStart, sValid;

    maskSh[t] = (t < 2) ? 1 : 0;
    hhat[t] = h_hat_g[b * DD + t];
    vtP[t] = vecs[t];
    vmP[t] = vecs[DD + t];
    phU[t] = vecs[2 * DD + t];
    if (t == 0) {
        sTm = Tmax_in[b]; sTmax0 = Tmax_in[b];
        sM = m_in[b]; sLast = 0; sStart = 1;
        sTotal = 0.f; sLogp = 0.f; sValid = 0;
    }
    __syncthreads();

    const float* P_b = P + (size_t)b * SS * NP;
    const float* dists_b = dists + (size_t)b * SS * SS;
    const float* d2d_b = d2depot + (size_t)b * SS;

    int h  = t >> 4;
    int sl = t & 15;

    for (int step = 0; step < NSTEPS; ++step) {
        // --- query (per-step GEMV hoisted to precompute) ------------------
        {
            float lastp = (step == 0) ? phU[t]
                                      : P_b[(size_t)sLast * NP + 2 * DD + t];
            q[t] = hhat[t] + lastp + sTm * vtP[t] + (float)sM * vmP[t];
        }
        __syncthreads();

        // --- compat[h][s] = q_h . gK[s,h] / 4, masked (b128 loads) --------
        for (int s8 = 0; s8 < 8; ++s8) {
            int s = s8 * 16 + sl;
            const float4* gk4 = (const float4*)(P_b + (size_t)s * NP + h * DKH);
            float cacc = 0.f;
#pragma unroll
            for (int d4 = 0; d4 < 4; ++d4) {
                float4 g = gk4[d4];
                int d = d4 * 4;
                cacc += q[h * DKH + d]     * g.x + q[h * DKH + d + 1] * g.y
                      + q[h * DKH + d + 2] * g.z + q[h * DKH + d + 3] * g.w;
            }
            att[h][s] = maskSh[s] ? NEGINF : (cacc * 0.25f);
        }
        __syncthreads();

        // --- softmax over s per head (16 threads per head) ----------------
        {
            float pm = NEGINF;
            for (int s8 = 0; s8 < 8; ++s8) pm = fmaxf(pm, att[h][s8 * 16 + sl]);
            red[t] = pm;
            __syncthreads();
            for (int off = 8; off > 0; off >>= 1) {
                if (sl < off) red[t] = fmaxf(red[t], red[t + off]);
                __syncthreads();
            }
            float hmax = red[h * 16];
            __syncthreads();
            float ps = 0.f;
            for (int s8 = 0; s8 < 8; ++s8) {
                int s = s8 * 16 + sl;
                float e = __expf(att[h][s] - hmax);
                att[h][s] = e;
                ps += e;
            }
            red[t] = ps;
            __syncthreads();
            for (int off = 8; off > 0; off >>= 1) {
                if (sl < off) red[t] += red[t + off];
                __syncthreads();
            }
            float inv = 1.f / red[h * 16];
            __syncthreads();
            for (int s8 = 0; s8 < 8; ++s8) att[h][s8 * 16 + sl] *= inv;
        }
        __syncthreads();

        // --- heads[h][d] = sum_s attn * gV --------------------------------
        {
            float acc = 0.f;
            for (int s = 0; s < SS; ++s)
                acc += att[h][s] * P_b[(size_t)s * NP + DD + h * DKH + sl];
            headsArr[h * DKH + sl] = acc;
        }
        __syncthreads();

        // --- logits[s] = 10*tanh((heads . M[s])/sqrt(128)), masked --------
        {
            const float4* lk4 = (const float4*)(P_b + (size_t)t * NP + 3 * DD);
            float acc = 0.f;
#pragma unroll 8
            for (int j4 = 0; j4 < 32; ++j4) {
                float4 g = lk4[j4];
                int j = j4 * 4;
                acc += headsArr[j] * g.x + headsArr[j + 1] * g.y
                     + headsArr[j + 2] * g.z + headsArr[j + 3] * g.w;
            }
            float lg = tanhf(acc * 0.08838834764831845f) * 10.f;
            logits[t] = maskSh[t] ? NEGINF : lg;
        }
        __syncthreads();

        // --- argmax (first-max tie-break) + logsumexp over 128 ------------
        red[t] = logits[t]; redi[t] = t;
        __syncthreads();
        for (int off = 64; off > 0; off >>= 1) {
            if (t < off) {
                float a = red[t], c2 = red[t + off];
                int ia = redi[t], ic = redi[t + off];
                if (c2 > a || (c2 == a && ic < ia)) { red[t] = c2; redi[t] = ic; }
            }
            __syncthreads();
        }
        float vmax = red[0];
        int city = redi[0];
        __syncthreads();
        red[t] = __expf(logits[t] - vmax);
        __syncthreads();
        for (int off = 64; off > 0; off >>= 1) {
            if (t < off) red[t] += red[t + off];
            __syncthreads();
        }
        float lse = logf(red[0]);
        __syncthreads();

        // --- scalar tour bookkeeping (thread 0) ---------------------------
        if (t == 0) {
            sLogp += logits[city] - vmax - lse; // log_softmax at chosen city
            float d_lc = dists_b[(size_t)sLast * SS + city];
            float d_cd = d2d_b[city];
            if (sStart) d_lc = d_cd;
            int is_valid = (d_lc + d_cd <= sTm) ? 1 : 0;
            if (is_valid) {
                sTm -= d_lc;
                if (sStart) sStart = 0;
                sTotal += scores[(size_t)b * SS + city];
            }
            sValid = is_valid;
        }
        __syncthreads();

        // --- all_g over s=1..127 (OLD last, OLD mask, NEW Tm) -------------
        int flag = 1;
        if (t >= 1) {
            float dlo = dists_b[(size_t)sLast * SS + t];
            flag = ((dlo + d2d_b[t] > sTm) || maskSh[t]) ? 1 : 0;
        }
        redi[t] = flag;
        __syncthreads();
        for (int off = 64; off > 0; off >>= 1) {
            if (t < off) redi[t] &= redi[t + off];
            __syncthreads();
        }
        if (t == 0) {
            int all_g = redi[0];
            int dbl = (all_g && (sM - 1 > 0)) ? 1 : 0;
            if (dbl) { sM -= 1; sStart = 1; sTm = sTmax0; }
            if (sValid) sLast = city;
            maskSh[city] = 1;
            out_sol[(size_t)b * NSTEPS + step] = (float)city;
        }
        __syncthreads();
    }

    if (t == 0) {
        out_logp[b] = sLogp;
        out_total[b] = sTotal;
    }
}

// ---------------------------------------------------------------------------
extern "C" void kernel_launch(void* const* d_in, const int* in_sizes, int n_in,
                              void* d_out, int out_size, void* d_ws, size_t ws_size,
                              hipStream_t stream) {
    const float* enc     = (const float*)d_in[0];
    const float* loc     = (const float*)d_in[1];
    const float* scores  = (const float*)d_in[2];
    const float* TmaxA   = (const float*)d_in[3];
    const int*   mA      = (const int*)d_in[4];
    const float* W_ctx   = (const float*)d_in[5];
    const float* W_upd   = (const float*)d_in[6];
    const float* W_nodes = (const float*)d_in[7];
    const float* W_out   = (const float*)d_in[8];
    const float* vtmax   = (const float*)d_in[9];
    const float* vm      = (const float*)d_in[10];
    const float* W_ph    = (const float*)d_in[11];

    float* ws    = (float*)d_ws;
    float* P     = ws;                                  // B*S*512 floats
    float* dists = P + (size_t)BB * SS * NP;            // B*S*S
    float* d2dep = dists + (size_t)BB * SS * SS;        // B*S
    float* hhat  = d2dep + (size_t)BB * SS;             // B*D
    float* Wbig  = hhat + (size_t)BB * DD;              // 128*512
    float* vecs  = Wbig + (size_t)DD * NP;              // 3*128

    float* out      = (float*)d_out;
    float* out_logp = out;            // [B]
    float* out_tot  = out + BB;       // [B]
    float* out_sol  = out + 2 * BB;   // [B, 126]

    wbig_kernel<<<DD, DD, 0, stream>>>(W_nodes, W_upd, W_out, Wbig);
    vecs_kernel<<<1, DD, 0, stream>>>(W_upd, vtmax, vm, W_ph, vecs);
    geom_kernel<<<BB, 128, 0, stream>>>(loc, dists, d2dep);
    hhat_kernel<<<BB, 128, 0, stream>>>(enc, W_ctx, hhat);

    dim3 g3((BB * SS) / 16, NP / 16 / 8); // (4096, 4), 8 waves/block
    proj_wmma_kernel<<<g3, 256, 0, stream>>>(enc, Wbig, P);

    decode_kernel<<<BB, 128, 0, stream>>>(scores, TmaxA, mA, P, vecs,
                                          dists, d2dep, hhat,
                                          out_logp, out_tot, out_sol);
}